// RecurrentNetv1_46394236731516
// MI455X (gfx1250) — compile-verified
//
#include <hip/hip_runtime.h>
#include <math.h>

// ---------------------------------------------------------------------------
// RecurrentNetv1 for MI455X (gfx1250, wave32, WMMA)
//
// Phase 1 (prep): zero-padded weight/bias copies into workspace so all WMMA
//                 fragment loads are unguarded b64 loads.
// Phase 2 (gru2_persistent): 8 workgroups x 512 threads; each owns 16 batch
//                 rows; fused input-path GEMM + 2-layer GRU recurrence with
//                 state in LDS; f32 WMMA (V_WMMA_F32_16X16X4_F32) for exact
//                 fp32 math on the latency-critical serial path.
// Phase 3 (mlp_head): 8192 workgroups; fused GEMM+LayerNorm+LeakyReLU x3
//                 plus mu/logvar heads and reparameterization, all in LDS.
//
// GEMM engine: two N-tiles per wave (independent WMMA chains), B-fragment
// software double-buffering (k+4 loads issued before k WMMAs), and
// global_prefetch of the next tile pair to cover L2 latency.
// ---------------------------------------------------------------------------

typedef __attribute__((ext_vector_type(2))) float v2f;
typedef __attribute__((ext_vector_type(8))) float v8f;

constexpr int B_   = 128;
constexpr int T_   = 1024;
constexpr int HIN_ = 129;   // input features
constexpr int NR_  = 200;   // GRU hidden
constexpr int NZ_  = 129;   // latent / output
constexpr int Hp   = 208;   // NR padded to 16
constexpr int Gp   = 608;   // 3*NR padded to 16
constexpr int KX   = 132;   // HIN padded to 4 (K granularity of wmma f32)
constexpr int NZp  = 144;   // NZ padded to 16

// ---- workspace layout (floats) --------------------------------------------
constexpr size_t SZ_WIH0 = (size_t)Gp * KX;     // 608x132
constexpr size_t SZ_WHH  = (size_t)Gp * Hp;     // 608x208
constexpr size_t SZ_W12  = (size_t)Hp * Hp;     // 208x208
constexpr size_t SZ_W3   = (size_t)NZp * Hp;    // 144x208
constexpr size_t SZ_WH   = (size_t)NZp * NZp;   // 144x144

constexpr size_t OFF_WIH0 = 0;
constexpr size_t OFF_WHH0 = OFF_WIH0 + SZ_WIH0;
constexpr size_t OFF_WIH1 = OFF_WHH0 + SZ_WHH;
constexpr size_t OFF_WHH1 = OFF_WIH1 + SZ_WHH;
constexpr size_t OFF_W1   = OFF_WHH1 + SZ_WHH;
constexpr size_t OFF_W2   = OFF_W1   + SZ_W12;
constexpr size_t OFF_W3   = OFF_W2   + SZ_W12;
constexpr size_t OFF_WMU  = OFF_W3   + SZ_W3;
constexpr size_t OFF_WLS  = OFF_WMU  + SZ_WH;
constexpr size_t OFF_BIH0 = OFF_WLS  + SZ_WH;
constexpr size_t OFF_BHH0 = OFF_BIH0 + Gp;
constexpr size_t OFF_BIH1 = OFF_BHH0 + Gp;
constexpr size_t OFF_BHH1 = OFF_BIH1 + Gp;
constexpr size_t OFF_B1   = OFF_BHH1 + Gp;
constexpr size_t OFF_B2   = OFF_B1   + Hp;
constexpr size_t OFF_B3   = OFF_B2   + Hp;
constexpr size_t OFF_BMU  = OFF_B3   + NZp;
constexpr size_t OFF_BLS  = OFF_BMU  + NZp;
constexpr size_t OFF_O1   = OFF_BLS  + NZp;
constexpr size_t SZ_O1    = (size_t)B_ * T_ * Hp;   // GRU output, padded

// GRU kernel dynamic LDS (floats): xA + h0 + h1 + xg + gg
constexpr int GRU_SMEM_F = 16 * KX + 2 * (16 * Hp) + 2 * (16 * Gp);
constexpr int GRU_THREADS = 512;   // 16 waves: 4/SIMD for latency hiding
constexpr int GRU_WAVES   = GRU_THREADS / 32;

// ---------------------------------------------------------------------------
// prep: zero-padded copy  dst[Np x Kp] <- src[N x K]
// ---------------------------------------------------------------------------
__global__ void pad2d(const float* __restrict__ src, float* __restrict__ dst,
                      int N, int K, int Np, int Kp) {
  int tot = Np * Kp;
  for (int i = blockIdx.x * blockDim.x + threadIdx.x; i < tot;
       i += gridDim.x * blockDim.x) {
    int n = i / Kp, k = i - n * Kp;
    dst[i] = (n < N && k < K) ? src[n * K + k] : 0.f;
  }
}

// ---------------------------------------------------------------------------
// 16xN GEMM tile engine:  Out[16 x 16*Ntiles] = A[16 x Kp] * Bw^T + bias
//   A   : LDS, row-major stride Kp (Kp % 4 == 0)       -> ds_load_b64 frags
//   Bw  : global, padded weights [Npad x Kp] (W[n][k]) -> global_load_b64
//   Out : LDS, row-major stride ldOut
// f32 WMMA 16x16x4: A-frag lane L = {A[m=L%16][k0+2*(L/16)+0,1]},
//                   B-frag lane L = {W[n0+L%16][k0+2*(L/16)+0,1]},
//                   D lane L vgpr r = D[m=r+8*(L/16)][n=n0+L%16].
// Two N-tiles per wave + B double-buffering: two independent WMMA chains,
// next-k loads issued before current-k WMMAs, next-pair prefetched.
// Requires EXEC all ones: all guards are wave-uniform.
// ---------------------------------------------------------------------------
__device__ __forceinline__ void gemm16(const float* __restrict__ A, int Kp,
                                       const float* __restrict__ Bw,
                                       const float* __restrict__ bias,
                                       float* __restrict__ Out, int ldOut,
                                       int Ntiles, int lane, int wave,
                                       int nwaves) {
  const int half = lane >> 4;       // K-pair select
  const int lm   = lane & 15;
  const float* arow = A + lm * Kp + 2 * half;
  for (int nt0 = wave; nt0 < Ntiles; nt0 += 2 * nwaves) {
    const int nt1  = nt0 + nwaves;
    const bool two = (nt1 < Ntiles);              // wave-uniform
    const int n0 = nt0 * 16 + lm;
    const int n1 = (two ? nt1 : nt0) * 16 + lm;   // dummy=tile0 in tail
    const float* brow0 = Bw + (size_t)n0 * Kp + 2 * half;
    const float* brow1 = Bw + (size_t)n1 * Kp + 2 * half;

    // prefetch next tile pair (speculative, warms L2->WGP path)
    const int ntp = nt0 + 2 * nwaves;
    if (ntp < Ntiles)
      __builtin_prefetch(Bw + (size_t)(ntp * 16 + lm) * Kp + 2 * half, 0, 1);

    v8f acc0 = {0.f, 0.f, 0.f, 0.f, 0.f, 0.f, 0.f, 0.f};
    v8f acc1 = {0.f, 0.f, 0.f, 0.f, 0.f, 0.f, 0.f, 0.f};
    v2f b0 = *(const v2f*)(brow0);
    v2f b1 = *(const v2f*)(brow1);
    int k0 = 0;
    for (; k0 < Kp - 4; k0 += 4) {
      // issue next-k fragment loads before consuming current-k fragments
      v2f nb0 = *(const v2f*)(brow0 + k0 + 4);
      v2f nb1 = *(const v2f*)(brow1 + k0 + 4);
      v2f a   = *(const v2f*)(arow + k0);
      acc0 = __builtin_amdgcn_wmma_f32_16x16x4_f32(false, a, false, b0,
                                                   (short)0, acc0, false, false);
      acc1 = __builtin_amdgcn_wmma_f32_16x16x4_f32(false, a, false, b1,
                                                   (short)0, acc1, false, false);
      b0 = nb0;
      b1 = nb1;
    }
    {
      v2f a = *(const v2f*)(arow + k0);
      acc0 = __builtin_amdgcn_wmma_f32_16x16x4_f32(false, a, false, b0,
                                                   (short)0, acc0, false, false);
      acc1 = __builtin_amdgcn_wmma_f32_16x16x4_f32(false, a, false, b1,
                                                   (short)0, acc1, false, false);
    }
    const float bv0 = bias[n0];
#pragma unroll
    for (int r = 0; r < 8; ++r)
      Out[(r + 8 * half) * ldOut + n0] = acc0[r] + bv0;
    if (two) {
      const float bv1 = bias[n1];
#pragma unroll
      for (int r = 0; r < 8; ++r)
        Out[(r + 8 * half) * ldOut + n1] = acc1[r] + bv1;
    }
  }
}

__device__ __forceinline__ float sigm(float x) {
  return 1.f / (1.f + __expf(-x));
}

// GRU gate math: h <- (1-z)*n + z*h  using xg (input path) and gg (hidden path)
__device__ __forceinline__ void gru_gates(const float* __restrict__ xg,
                                          const float* __restrict__ gg,
                                          float* __restrict__ h, int tid,
                                          int nthreads) {
  for (int i = tid; i < 16 * NR_; i += nthreads) {
    int m = i / NR_, g = i - m * NR_;
    const float* xr = xg + m * Gp;
    const float* hr = gg + m * Gp;
    float r = sigm(xr[g] + hr[g]);
    float z = sigm(xr[NR_ + g] + hr[NR_ + g]);
    float n = tanhf(xr[2 * NR_ + g] + r * hr[2 * NR_ + g]);
    float* hm = h + m * Hp;
    hm[g] = (1.f - z) * n + z * hm[g];
  }
}

// ---------------------------------------------------------------------------
// Fused 2-layer GRU, persistent over T. 8 blocks x 512 threads (16 waves).
// Block owns batch rows [16*blockIdx, +16). State h0/h1 lives in LDS.
// Weights stream from L2 (padded copies in ws). Writes o1 [B*T x Hp] to ws.
// ---------------------------------------------------------------------------
__global__ void __launch_bounds__(GRU_THREADS) gru2_persistent(
    const float* __restrict__ x, float* __restrict__ ws) {
  extern __shared__ float smem[];
  float* xA = smem;                 // 16 x 132 staged input tile (zero-padded)
  float* h0 = xA + 16 * KX;         // 16 x 208
  float* h1 = h0 + 16 * Hp;         // 16 x 208
  float* xg = h1 + 16 * Hp;         // 16 x 608 input-path gates
  float* gg = xg + 16 * Gp;         // 16 x 608 hidden-path gates

  const int tid = threadIdx.x, lane = tid & 31, wave = tid >> 5;
  const int b0 = blockIdx.x * 16;

  const float* Wih0 = ws + OFF_WIH0;
  const float* Whh0 = ws + OFF_WHH0;
  const float* Wih1 = ws + OFF_WIH1;
  const float* Whh1 = ws + OFF_WHH1;
  const float* bih0 = ws + OFF_BIH0;
  const float* bhh0 = ws + OFF_BHH0;
  const float* bih1 = ws + OFF_BIH1;
  const float* bhh1 = ws + OFF_BHH1;
  float* o1 = ws + OFF_O1;

  for (int i = tid; i < GRU_SMEM_F; i += GRU_THREADS) smem[i] = 0.f;
  __syncthreads();

  for (int t = 0; t < T_; ++t) {
    // stage x[b0..b0+15, t, :] -> xA (zero-padded K)
    for (int i = tid; i < 16 * KX; i += GRU_THREADS) {
      int m = i / KX, k = i - m * KX;
      xA[i] = (k < HIN_)
                  ? x[((size_t)(b0 + m) * T_ + t) * HIN_ + k]
                  : 0.f;
    }
    __syncthreads();

    // layer 0: xg = xA*Wih0^T + bih0 ; gg = h0*Whh0^T + bhh0
    gemm16(xA, KX, Wih0, bih0, xg, Gp, Gp / 16, lane, wave, GRU_WAVES);
    gemm16(h0, Hp, Whh0, bhh0, gg, Gp, Gp / 16, lane, wave, GRU_WAVES);
    __syncthreads();
    gru_gates(xg, gg, h0, tid, GRU_THREADS);
    __syncthreads();

    // layer 1: xg = h0*Wih1^T + bih1 ; gg = h1*Whh1^T + bhh1
    gemm16(h0, Hp, Wih1, bih1, xg, Gp, Gp / 16, lane, wave, GRU_WAVES);
    gemm16(h1, Hp, Whh1, bhh1, gg, Gp, Gp / 16, lane, wave, GRU_WAVES);
    __syncthreads();
    gru_gates(xg, gg, h1, tid, GRU_THREADS);
    __syncthreads();

    // spill h1 -> o1 (padded cols are already zero)
    for (int i = tid; i < 16 * Hp; i += GRU_THREADS) {
      int m = i / Hp, c = i - m * Hp;
      o1[((size_t)(b0 + m) * T_ + t) * Hp + c] = h1[i];
    }
    __syncthreads();
  }
}

// ---------------------------------------------------------------------------
// One MLP layer: Out = LeakyReLU(LayerNorm(A*W^T + b)), all in LDS.
// ---------------------------------------------------------------------------
__device__ __forceinline__ void mlp_layer(
    const float* __restrict__ Ain, float* __restrict__ Outb,
    const float* __restrict__ Wp, const float* __restrict__ bp,
    const float* __restrict__ g, const float* __restrict__ bb,
    int Ntiles, int Kp, int Nvalid,
    float* __restrict__ red1, float* __restrict__ red2, float* __restrict__ mv,
    int tid, int lane, int wave) {
  gemm16(Ain, Kp, Wp, bp, Outb, Hp, Ntiles, lane, wave, 8);
  __syncthreads();

  // LayerNorm stats: 16 threads per row
  const int row = tid >> 4, c0 = tid & 15;
  float s = 0.f, s2 = 0.f;
  for (int c = c0; c < Nvalid; c += 16) {
    float v = Outb[row * Hp + c];
    s += v;
    s2 += v * v;
  }
  red1[tid] = s;
  red2[tid] = s2;
  __syncthreads();
  if (c0 == 0) {
    float a = 0.f, q = 0.f;
    for (int j = 0; j < 16; ++j) {
      a += red1[row * 16 + j];
      q += red2[row * 16 + j];
    }
    float m = a / (float)Nvalid;
    float var = q / (float)Nvalid - m * m;
    mv[row * 2] = m;
    mv[row * 2 + 1] = rsqrtf(var + 1e-5f);
  }
  __syncthreads();
  for (int i = tid; i < 16 * Hp; i += 256) {
    int r = i / Hp, c = i - r * Hp;
    float v = 0.f;
    if (c < Nvalid) {
      v = (Outb[i] - mv[r * 2]) * mv[r * 2 + 1] * g[c] + bb[c];
      v = v > 0.f ? v : 0.1f * v;
    }
    Outb[i] = v;
  }
  __syncthreads();
}

// ---------------------------------------------------------------------------
// MLP + heads + reparameterization. One block = 16 (b,t) positions.
// ---------------------------------------------------------------------------
__global__ void __launch_bounds__(256) mlp_head(
    const float* __restrict__ ws,
    const float* __restrict__ ln1g, const float* __restrict__ ln1b,
    const float* __restrict__ ln2g, const float* __restrict__ ln2b,
    const float* __restrict__ ln3g, const float* __restrict__ ln3b,
    const float* __restrict__ noise, float* __restrict__ out) {
  __shared__ float bufA[16 * Hp];
  __shared__ float bufB[16 * Hp];
  __shared__ float red1[256], red2[256], mv[32];

  const int tid = threadIdx.x, lane = tid & 31, wave = tid >> 5;
  const size_t p0 = (size_t)blockIdx.x * 16;
  const float* o1 = ws + OFF_O1;

  for (int i = tid; i < 16 * Hp; i += 256)
    bufA[i] = o1[(p0 + i / Hp) * Hp + (i % Hp)];
  __syncthreads();

  mlp_layer(bufA, bufB, ws + OFF_W1, ws + OFF_B1, ln1g, ln1b,
            Hp / 16, Hp, NR_, red1, red2, mv, tid, lane, wave);
  mlp_layer(bufB, bufA, ws + OFF_W2, ws + OFF_B2, ln2g, ln2b,
            Hp / 16, Hp, NR_, red1, red2, mv, tid, lane, wave);
  mlp_layer(bufA, bufB, ws + OFF_W3, ws + OFF_B3, ln3g, ln3b,
            NZp / 16, Hp, NZ_, red1, red2, mv, tid, lane, wave);

  // heads: mu and logvar accumulators share fragment coordinates, so fuse
  // the reparameterization at store time. Two independent WMMA chains.
  const float* Wmu = ws + OFF_WMU;
  const float* Wls = ws + OFF_WLS;
  const float* bmu = ws + OFF_BMU;
  const float* bls = ws + OFF_BLS;
  const int half = lane >> 4, lm = lane & 15;
  const float* arow = bufB + lm * Hp + 2 * half;
  for (int nt = wave; nt < NZp / 16; nt += 8) {
    const int n = nt * 16 + lm;
    const float* mrow = Wmu + (size_t)n * NZp + 2 * half;
    const float* lrow = Wls + (size_t)n * NZp + 2 * half;
    v8f am = {0.f, 0.f, 0.f, 0.f, 0.f, 0.f, 0.f, 0.f};
    v8f al = {0.f, 0.f, 0.f, 0.f, 0.f, 0.f, 0.f, 0.f};
    for (int k0 = 0; k0 < NZp; k0 += 4) {
      v2f a  = *(const v2f*)(arow + k0);
      v2f wm = *(const v2f*)(mrow + k0);
      v2f wl = *(const v2f*)(lrow + k0);
      am = __builtin_amdgcn_wmma_f32_16x16x4_f32(false, a, false, wm,
                                                 (short)0, am, false, false);
      al = __builtin_amdgcn_wmma_f32_16x16x4_f32(false, a, false, wl,
                                                 (short)0, al, false, false);
    }
    if (n < NZ_) {
      float bmv = bmu[n], blv = bls[n];
#pragma unroll
      for (int r = 0; r < 8; ++r) {
        size_t p = p0 + (size_t)(r + 8 * half);
        float mu = am[r] + bmv;
        float lv = al[r] + blv;
        out[p * NZ_ + n] = mu + __expf(lv) * noise[p * NZ_ + n];
      }
    }
  }
}

// ---------------------------------------------------------------------------
extern "C" void kernel_launch(void* const* d_in, const int* in_sizes, int n_in,
                              void* d_out, int out_size, void* d_ws,
                              size_t ws_size, hipStream_t stream) {
  (void)in_sizes; (void)n_in; (void)out_size; (void)ws_size;
  const float* x     = (const float*)d_in[0];
  const float* noise = (const float*)d_in[1];
  float* ws  = (float*)d_ws;
  float* out = (float*)d_out;

  auto pad = [&](const void* src, size_t off, int N, int K, int Np, int Kp) {
    int tot = Np * Kp;
    int blocks = (tot + 255) / 256;
    if (blocks > 1024) blocks = 1024;
    pad2d<<<dim3(blocks), dim3(256), 0, stream>>>((const float*)src, ws + off,
                                                  N, K, Np, Kp);
  };

  // weights
  pad(d_in[2],  OFF_WIH0, 3 * NR_, HIN_, Gp,  KX);
  pad(d_in[3],  OFF_WHH0, 3 * NR_, NR_,  Gp,  Hp);
  pad(d_in[6],  OFF_WIH1, 3 * NR_, NR_,  Gp,  Hp);
  pad(d_in[7],  OFF_WHH1, 3 * NR_, NR_,  Gp,  Hp);
  pad(d_in[10], OFF_W1,   NR_,     NR_,  Hp,  Hp);
  pad(d_in[14], OFF_W2,   NR_,     NR_,  Hp,  Hp);
  pad(d_in[18], OFF_W3,   NZ_,     NR_,  NZp, Hp);
  pad(d_in[22], OFF_WMU,  NZ_,     NZ_,  NZp, NZp);
  pad(d_in[24], OFF_WLS,  NZ_,     NZ_,  NZp, NZp);
  // biases (as Np x 1)
  pad(d_in[4],  OFF_BIH0, 3 * NR_, 1, Gp,  1);
  pad(d_in[5],  OFF_BHH0, 3 * NR_, 1, Gp,  1);
  pad(d_in[8],  OFF_BIH1, 3 * NR_, 1, Gp,  1);
  pad(d_in[9],  OFF_BHH1, 3 * NR_, 1, Gp,  1);
  pad(d_in[11], OFF_B1,   NR_,     1, Hp,  1);
  pad(d_in[15], OFF_B2,   NR_,     1, Hp,  1);
  pad(d_in[19], OFF_B3,   NZ_,     1, NZp, 1);
  pad(d_in[23], OFF_BMU,  NZ_,     1, NZp, 1);
  pad(d_in[25], OFF_BLS,  NZ_,     1, NZp, 1);

  // sequential phase: 8 persistent blocks (one 16-row batch slice each)
  gru2_persistent<<<dim3(B_ / 16), dim3(GRU_THREADS),
                    GRU_SMEM_F * sizeof(float), stream>>>(x, ws);

  // parallel phase: one block per 16 (b,t) positions
  mlp_head<<<dim3((B_ * T_) / 16), dim3(256), 0, stream>>>(
      ws,
      (const float*)d_in[12], (const float*)d_in[13],
      (const float*)d_in[16], (const float*)d_in[17],
      (const float*)d_in[20], (const float*)d_in[21],
      noise, out);
}